// LinearGraphTransformer_12970801234192
// MI455X (gfx1250) — compile-verified
//
#include <hip/hip_runtime.h>
#include <math.h>
#include <stdint.h>

// MI455X / gfx1250, wave32. fp32 WMMA (v_wmma_f32_16x16x4_f32) + TDM
// (tensor_load_to_lds) async staging into LDS.
typedef __attribute__((ext_vector_type(2))) float v2f;
typedef __attribute__((ext_vector_type(8))) float v8f;
typedef __attribute__((ext_vector_type(4))) unsigned int u32x4;
typedef __attribute__((ext_vector_type(8))) int i32x8;

#define NR   100000
#define HD   512
#define ED   8
#define EPSV 1e-12f

// ---- workspace layout (float offsets) ----
static const size_t OFF_Q     = 0;
static const size_t OFF_K     = (size_t)NR * HD;
static const size_t OFF_V     = 2ull * NR * HD;
static const size_t OFF_KVP   = 3ull * NR * HD;             // 25*262144 partials
static const size_t OFF_KV    = OFF_KVP + 25ull * HD * HD;  // 262144
static const size_t OFF_KSUMP = OFF_KV + (size_t)HD * HD;   // 200*512
static const size_t OFF_QSQP  = OFF_KSUMP + 200ull * HD;    // 200
static const size_t OFF_KSQP  = OFF_QSQP + 200;             // 200
static const size_t OFF_KSUM  = OFF_KSQP + 200;             // 512
static const size_t OFF_SCAL  = OFF_KSUM + HD;              // 8
static const size_t OFF_DOT   = OFF_SCAL + 8;               // 100,000

static __device__ __forceinline__ v8f vzero8() {
  v8f z;
#pragma unroll
  for (int i = 0; i < 8; i++) z[i] = 0.f;
  return z;
}

// D = A(16x4) * B(4x16) + C, fp32.  A/B = v2f per lane, C/D = v8f per lane.
static __device__ __forceinline__ v8f wmma4(v2f a, v2f b, v8f c) {
  return __builtin_amdgcn_wmma_f32_16x16x4_f32(false, a, false, b, (short)0, c,
                                               false, false);
}

// Tensor Data Mover: DMA a [height x width] fp32 tile (row stride
// row_stride_elems) from global into contiguous LDS at lds_ptr.
// D# per CDNA5 ISA 8.3/8.4: group0 = {count/type/addr}, group1 = dims.
// Issue from ONE wave only (per-wave descriptor op, EXEC ignored); completion
// via TENSORcnt.
static __device__ __forceinline__ void tdm_load_2d(const float* gptr,
                                                   const float* lds_ptr,
                                                   int width, int height,
                                                   int row_stride_elems) {
  uint64_t ga = (uint64_t)(uintptr_t)gptr;
  uint32_t lo = (uint32_t)(uintptr_t)lds_ptr;  // LDS byte offset = addr[31:0]
  u32x4 g0;
  g0[0] = 1u;                                   // count=1 (valid), no gather
  g0[1] = lo;                                   // lds_addr
  g0[2] = (uint32_t)ga;                         // global_addr[31:0]
  g0[3] = ((uint32_t)(ga >> 32) & 0x01FFFFFFu)  // global_addr[56:32]
          | (2u << 30);                         // type=2 ("image")
  i32x8 g1;
  g1[0] = (int)(2u << 16);                      // wg_mask=0, data_size=4B
  g1[1] = (int)(((uint32_t)width & 0xFFFFu) << 16);   // tensor_dim0[15:0]
  g1[2] = (int)((((uint32_t)width) >> 16) |
                (((uint32_t)height & 0xFFFFu) << 16));// dim0[31:16]|dim1[15:0]
  g1[3] = (int)((((uint32_t)height) >> 16) |
                (((uint32_t)width & 0xFFFFu) << 16)); // dim1[31:16]|tile_dim0
  g1[4] = (int)((uint32_t)height & 0xFFFFu);    // tile_dim1 (tile_dim2=0)
  g1[5] = row_stride_elems;                     // tensor_dim0_stride[31:0]
  g1[6] = 0;                                    // stride hi, dim1_stride lo
  g1[7] = 0;
  asm volatile("tensor_load_to_lds %0, %1" ::"s"(g0), "s"(g1) : "memory");
}

// ---------------- K1: q,k,v = x @ W^T + b (raw, unnormalized) ---------------
// grid 3125 x 256 thr. 32 rows of x TDM-staged in 64KB LDS; each wave takes
// 12 (matrix, col-tile) jobs; two 16-row M-tiles share each B fragment.
__global__ __launch_bounds__(256) void lgt_qkv(
    const float* __restrict__ x,
    const float* __restrict__ Wq, const float* __restrict__ bq,
    const float* __restrict__ Wk, const float* __restrict__ bk,
    const float* __restrict__ Wv, const float* __restrict__ bv,
    float* __restrict__ q, float* __restrict__ k, float* __restrict__ v) {
  __shared__ float xs[32 * HD];
  const int r0 = blockIdx.x * 32;
  if (threadIdx.x < 32) {  // wave 0 issues the DMA
    tdm_load_2d(x + (size_t)r0 * HD, xs, 32 * HD, 1, 32 * HD);
    __builtin_amdgcn_s_wait_tensorcnt(0);
  }
  __syncthreads();
  const int wave = threadIdx.x >> 5;
  const int lane = threadIdx.x & 31;
  const int half = lane >> 4;
  const int ln = lane & 15;
  for (int job = wave; job < 96; job += 8) {
    const int mat = job >> 5;  // 0=q 1=k 2=v
    const int ct = job & 31;   // column tile
    const int cb = ct * 16;
    const float* W;
    const float* B;
    float* O;
    if (mat == 0) { W = Wq; B = bq; O = q; }
    else if (mat == 1) { W = Wk; B = bk; O = k; }
    else { W = Wv; B = bv; O = v; }
    v8f acc0 = vzero8();
    v8f acc1 = vzero8();
    // B[kk,n] = W[cb+n, k0+kk]  (contiguous K-pair per lane)
    const float* wrow = W + (size_t)(cb + ln) * HD + 2 * half;
    const float* a0p = xs + (size_t)ln * HD + 2 * half;
    const float* a1p = xs + (size_t)(16 + ln) * HD + 2 * half;
#pragma unroll 4
    for (int k0 = 0; k0 < HD; k0 += 4) {
      v2f bf = *(const v2f*)(wrow + k0);
      v2f a0 = *(const v2f*)(a0p + k0);
      v2f a1 = *(const v2f*)(a1p + k0);
      acc0 = wmma4(a0, bf, acc0);
      acc1 = wmma4(a1, bf, acc1);
    }
    const float bias = B[cb + ln];
#pragma unroll
    for (int r = 0; r < 8; r++) {
      const int rowA = r0 + r + half * 8;
      O[(size_t)rowA * HD + cb + ln] = acc0[r] + bias;
      O[(size_t)(rowA + 16) * HD + cb + ln] = acc1[r] + bias;
    }
  }
}

// ------------- K2: per-block partials of ||q||^2, ||k||^2, colsum(k) --------
__global__ __launch_bounds__(256) void lgt_partred(
    const float* __restrict__ q, const float* __restrict__ k,
    float* __restrict__ ksum_p, float* __restrict__ qsq_p,
    float* __restrict__ ksq_p) {
  const int b = blockIdx.x;   // 200 blocks x 500 rows
  const int t = threadIdx.x;  // 256
  const int rbeg = b * 500;
  float s0 = 0.f, s1 = 0.f, qs = 0.f, kq = 0.f;
  for (int r = 0; r < 500; r++) {
    const float* qr = q + (size_t)(rbeg + r) * HD;
    const float* kr = k + (size_t)(rbeg + r) * HD;
    float k0v = kr[t], k1v = kr[t + 256];
    float q0v = qr[t], q1v = qr[t + 256];
    s0 += k0v;
    s1 += k1v;
    qs += q0v * q0v + q1v * q1v;
    kq += k0v * k0v + k1v * k1v;
  }
  ksum_p[(size_t)b * HD + t] = s0;
  ksum_p[(size_t)b * HD + t + 256] = s1;
  __shared__ float red[256];
  red[t] = qs;
  __syncthreads();
  for (int s = 128; s > 0; s >>= 1) {
    if (t < s) red[t] += red[t + s];
    __syncthreads();
  }
  if (t == 0) qsq_p[b] = red[0];
  __syncthreads();
  red[t] = kq;
  __syncthreads();
  for (int s = 128; s > 0; s >>= 1) {
    if (t < s) red[t] += red[t + s];
    __syncthreads();
  }
  if (t == 0) ksq_p[b] = red[0];
}

// ------------- K3: fold partials -> ksum[512], invscale (deterministic) ----
__global__ void lgt_finalize(const float* __restrict__ ksum_p,
                             const float* __restrict__ qsq_p,
                             const float* __restrict__ ksq_p,
                             float* __restrict__ ksum,
                             float* __restrict__ scal) {
  const int t = threadIdx.x;  // 512
  float s = 0.f;
  for (int b = 0; b < 200; b++) s += ksum_p[(size_t)b * HD + t];
  ksum[t] = s;
  if (t == 0) {
    float qs = 0.f, kq = 0.f;
    for (int b = 0; b < 200; b++) { qs += qsq_p[b]; kq += ksq_p[b]; }
    float nq = fmaxf(sqrtf(qs), EPSV);
    float nk = fmaxf(sqrtf(kq), EPSV);
    scal[0] = 1.0f / ((float)NR * nq * nk);
  }
}

// ------------- K4: kv partials = k^T v, 128x128 macro tile, split-K --------
// grid (16, 25): 4x4 macro tiles x 25 row-splits of 4000 rows. 8 waves, each
// owns one 16-wide N strip, 8 M sub-tiles. Double-buffered TDM staging:
// wave 0 DMAs chunk c+1 into the alternate LDS buffer while all waves run
// WMMAs on chunk c; s_wait_tensorcnt + barrier flips the buffer.
__global__ __launch_bounds__(256) void lgt_kvpart(
    const float* __restrict__ k, const float* __restrict__ v,
    float* __restrict__ kvp) {
  __shared__ float ks[2 * 32 * 128];
  __shared__ float vs[2 * 32 * 128];
  const int tileM = blockIdx.x & 3;
  const int tileN = blockIdx.x >> 2;
  const int split = blockIdx.y;  // 25
  const int wave0 = (threadIdx.x < 32);
  const int lane = threadIdx.x & 31;
  const int wave = threadIdx.x >> 5;
  const int half = lane >> 4;
  const int ln = lane & 15;
  v8f acc[8];
#pragma unroll
  for (int m = 0; m < 8; m++) acc[m] = vzero8();
  const int rbase = split * 4000;
  if (wave0) {
    tdm_load_2d(k + (size_t)rbase * HD + tileM * 128, ks, 128, 32, HD);
    tdm_load_2d(v + (size_t)rbase * HD + tileN * 128, vs, 128, 32, HD);
    __builtin_amdgcn_s_wait_tensorcnt(0);
  }
  __syncthreads();
  int buf = 0;
  for (int c = 0; c < 125; c++) {
    if ((c + 1 < 125) && wave0) {  // prefetch next chunk into other buffer
      const int ib = rbase + (c + 1) * 32;
      tdm_load_2d(k + (size_t)ib * HD + tileM * 128,
                  ks + (buf ^ 1) * (32 * 128), 128, 32, HD);
      tdm_load_2d(v + (size_t)ib * HD + tileN * 128,
                  vs + (buf ^ 1) * (32 * 128), 128, 32, HD);
    }
    const float* ksb = ks + buf * (32 * 128);
    const float* vsb = vs + buf * (32 * 128);
#pragma unroll
    for (int k0 = 0; k0 < 32; k0 += 4) {
      // B[kk,n] = v[ib+k0+kk, nb+n]
      v2f bf;
      bf.x = vsb[(k0 + 2 * half) * 128 + wave * 16 + ln];
      bf.y = vsb[(k0 + 2 * half + 1) * 128 + wave * 16 + ln];
#pragma unroll
      for (int m = 0; m < 8; m++) {
        // A[mm,kk] = k[ib+k0+kk, mb+mm]  (k^T)
        v2f af;
        af.x = ksb[(k0 + 2 * half) * 128 + m * 16 + ln];
        af.y = ksb[(k0 + 2 * half + 1) * 128 + m * 16 + ln];
        acc[m] = wmma4(af, bf, acc[m]);
      }
    }
    if (wave0) __builtin_amdgcn_s_wait_tensorcnt(0);
    __syncthreads();
    buf ^= 1;
  }
  float* base = kvp + (size_t)split * (HD * HD);
#pragma unroll
  for (int m = 0; m < 8; m++) {
#pragma unroll
    for (int r = 0; r < 8; r++) {
      const int row = tileM * 128 + m * 16 + r + half * 8;
      const int col = tileN * 128 + wave * 16 + ln;
      base[(size_t)row * HD + col] = acc[m][r];
    }
  }
}

// ------------- K5: reduce kv partials (fixed order) ------------------------
__global__ void lgt_kvred(const float* __restrict__ kvp,
                          float* __restrict__ kv) {
  const int idx = blockIdx.x * 256 + threadIdx.x;  // 262144
  float s = 0.f;
  for (int p = 0; p < 25; p++) s += kvp[(size_t)p * (HD * HD) + idx];
  kv[idx] = s;
}

// ------------- K5b: dot[i] = q_raw[i] . ksum_raw ---------------------------
__global__ __launch_bounds__(256) void lgt_dot(const float* __restrict__ q,
                                               const float* __restrict__ ksum,
                                               float* __restrict__ dotv) {
  const int g = threadIdx.x >> 3;  // row group of 8 lanes
  const int j = threadIdx.x & 7;
  const int row = blockIdx.x * 32 + g;
  const float* qr = q + (size_t)row * HD;
  float s = 0.f;
  for (int c = j; c < HD; c += 8) s += qr[c] * ksum[c];
  s += __shfl_xor(s, 4, 8);
  s += __shfl_xor(s, 2, 8);
  s += __shfl_xor(s, 1, 8);
  if (j == 0) dotv[row] = s;
}

// ------------- K6: repr = (v + (q@kv)*invscale)/max(1+dot*invscale, eps) ---
__global__ __launch_bounds__(256) void lgt_out(
    const float* __restrict__ q, const float* __restrict__ v,
    const float* __restrict__ kv, const float* __restrict__ dotv,
    const float* __restrict__ scal, float* __restrict__ outR) {
  __shared__ float qsm[32 * HD];
  const int r0 = blockIdx.x * 32;
  if (threadIdx.x < 32) {
    tdm_load_2d(q + (size_t)r0 * HD, qsm, 32 * HD, 1, 32 * HD);
    __builtin_amdgcn_s_wait_tensorcnt(0);
  }
  __syncthreads();
  const float sc = scal[0];
  const int wave = threadIdx.x >> 5;
  const int lane = threadIdx.x & 31;
  const int half = lane >> 4;
  const int ln = lane & 15;
  for (int ct = wave; ct < 32; ct += 8) {
    const int cb = ct * 16;
    v8f acc0 = vzero8();
    v8f acc1 = vzero8();
    const float* a0p = qsm + (size_t)ln * HD + 2 * half;
    const float* a1p = qsm + (size_t)(16 + ln) * HD + 2 * half;
#pragma unroll 4
    for (int k0 = 0; k0 < HD; k0 += 4) {
      // B[kk,n] = kv[k0+kk, cb+n] (no transpose)
      v2f bf;
      bf.x = kv[(size_t)(k0 + 2 * half) * HD + cb + ln];
      bf.y = kv[(size_t)(k0 + 2 * half + 1) * HD + cb + ln];
      v2f a0 = *(const v2f*)(a0p + k0);
      v2f a1 = *(const v2f*)(a1p + k0);
      acc0 = wmma4(a0, bf, acc0);
      acc1 = wmma4(a1, bf, acc1);
    }
#pragma unroll
    for (int r = 0; r < 8; r++) {
      const int rowA = r0 + r + half * 8;
      {
        float at = 1.f + dotv[rowA] * sc;
        float den = fmaxf(at, EPSV);
        size_t o = (size_t)rowA * HD + cb + ln;
        outR[o] = (v[o] + acc0[r] * sc) / den;
      }
      {
        const int rowB = rowA + 16;
        float at = 1.f + dotv[rowB] * sc;
        float den = fmaxf(at, EPSV);
        size_t o = (size_t)rowB * HD + cb + ln;
        outR[o] = (v[o] + acc1[r] * sc) / den;
      }
    }
  }
}

// ------------- K7: logits = repr @ We^T + be (E=8, VALU) -------------------
__global__ __launch_bounds__(256) void lgt_logits(
    const float* __restrict__ repr, const float* __restrict__ We,
    const float* __restrict__ be, float* __restrict__ outL) {
  __shared__ float rs[32 * HD];
  const int r0 = blockIdx.x * 32;
  if (threadIdx.x < 32) {
    tdm_load_2d(repr + (size_t)r0 * HD, rs, 32 * HD, 1, 32 * HD);
    __builtin_amdgcn_s_wait_tensorcnt(0);
  }
  __syncthreads();
  const int g = threadIdx.x >> 3;  // 0..31 rows
  const int e = threadIdx.x & 7;
  const float* wr = We + (size_t)e * HD;
  const float* rrow = rs + (size_t)g * HD;
  float s = be[e];
  for (int c = 0; c < HD; c++) s += rrow[c] * wr[c];
  outL[(size_t)(r0 + g) * ED + e] = s;
}

extern "C" void kernel_launch(void* const* d_in, const int* in_sizes, int n_in,
                              void* d_out, int out_size, void* d_ws,
                              size_t ws_size, hipStream_t stream) {
  (void)in_sizes; (void)n_in; (void)out_size; (void)ws_size;
  const float* x = (const float*)d_in[0];
  const float* Wq = (const float*)d_in[1];
  const float* bq = (const float*)d_in[2];
  const float* Wk = (const float*)d_in[3];
  const float* bk = (const float*)d_in[4];
  const float* Wv = (const float*)d_in[5];
  const float* bv = (const float*)d_in[6];
  const float* We = (const float*)d_in[7];
  const float* be = (const float*)d_in[8];

  float* ws = (float*)d_ws;
  float* q = ws + OFF_Q;
  float* k = ws + OFF_K;
  float* v = ws + OFF_V;
  float* kvp = ws + OFF_KVP;
  float* kv = ws + OFF_KV;
  float* ksum_p = ws + OFF_KSUMP;
  float* qsq_p = ws + OFF_QSQP;
  float* ksq_p = ws + OFF_KSQP;
  float* ksum = ws + OFF_KSUM;
  float* scal = ws + OFF_SCAL;
  float* dotv = ws + OFF_DOT;

  float* outR = (float*)d_out;
  float* outL = outR + (size_t)NR * HD;

  lgt_qkv<<<NR / 32, 256, 0, stream>>>(x, Wq, bq, Wk, bk, Wv, bv, q, k, v);
  lgt_partred<<<200, 256, 0, stream>>>(q, k, ksum_p, qsq_p, ksq_p);
  lgt_finalize<<<1, 512, 0, stream>>>(ksum_p, qsq_p, ksq_p, ksum, scal);
  lgt_kvpart<<<dim3(16, 25), 256, 0, stream>>>(k, v, kvp);
  lgt_kvred<<<(HD * HD) / 256, 256, 0, stream>>>(kvp, kv);
  lgt_dot<<<NR / 32, 256, 0, stream>>>(q, ksum, dotv);
  lgt_out<<<NR / 32, 256, 0, stream>>>(q, v, kv, dotv, scal, outR);
  lgt_logits<<<NR / 32, 256, 0, stream>>>(outR, We, be, outL);
}